// SAGE_7086696038727
// MI455X (gfx1250) — compile-verified
//
#include <hip/hip_runtime.h>
#include <hip/hip_bf16.h>

#define GN 100000
#define GE 1600000
#define DIN 128
#define DHID 128
#define DOUT 64

typedef __attribute__((ext_vector_type(2))) float v2f;
typedef __attribute__((ext_vector_type(8))) float v8f;

// ---------------------------------------------------------------------------
// Zero a float buffer (workspace is poisoned 0xAA by the harness; we must
// clear accumulators every launch for determinism).
__global__ void sage_fzero(float* __restrict__ p, int n) {
    int i = blockIdx.x * blockDim.x + threadIdx.x;
    if (i < n) p[i] = 0.0f;
}

// ---------------------------------------------------------------------------
// deg[v] = number of incoming edges (float, reused by both layers).
__global__ void sage_deg(const int* __restrict__ dst, float* __restrict__ deg, int E) {
    int e = blockIdx.x * blockDim.x + threadIdx.x;
    if (e < E) unsafeAtomicAdd(&deg[dst[e]], 1.0f);
}

// ---------------------------------------------------------------------------
// One wave per edge: lane l moves 4 contiguous floats (float4 load, 4 hw fp32
// atomics). Both h (51.2MB) and agg (51.2MB) fit in the 192MB L2, so these
// atomics are L2-resident.
__global__ void sage_scatter(const float* __restrict__ h,
                             const int* __restrict__ src,
                             const int* __restrict__ dst,
                             float* __restrict__ agg, int E) {
    int wave = (blockIdx.x * blockDim.x + threadIdx.x) >> 5;
    int lane = threadIdx.x & 31;
    if (wave >= E) return;
    int s = src[wave];
    int d = dst[wave];
    const float4 v = *(const float4*)(h + (size_t)s * 128 + lane * 4);
    float* o = agg + (size_t)d * 128 + lane * 4;
    unsafeAtomicAdd(o + 0, v.x);
    unsafeAtomicAdd(o + 1, v.y);
    unsafeAtomicAdd(o + 2, v.z);
    unsafeAtomicAdd(o + 3, v.w);
}

// ---------------------------------------------------------------------------
// agg[i] = (agg[i] + h[i]) / (deg[row] + 1)   (feature dim is 128 both layers)
__global__ void sage_neigh(float* __restrict__ agg, const float* __restrict__ h,
                           const float* __restrict__ deg, int total) {
    int i = blockIdx.x * blockDim.x + threadIdx.x;
    if (i < total) {
        float r = 1.0f / (deg[i >> 7] + 1.0f);
        agg[i] = (agg[i] + h[i]) * r;
    }
}

// ---------------------------------------------------------------------------
// fp32 WMMA GEMM:  C[M=16*grid][NCOLS] = A[M][128] * B[128][NCOLS] + bias
// One wave per 16x16 output tile; blockDim = 32 * (NCOLS/16); grid = M/16.
// K loop: 32 x V_WMMA_F32_16X16X4_F32.
//
// Fragment layouts per CDNA5 ISA 7.12.2 (wave32):
//   A 16x4 : lane = l16 + 16*half holds row M=l16, VGPR0=K=2*half, VGPR1=K=2*half+1
//   B 4x16 : lane holds col N=l16,  VGPR0=K=2*half, VGPR1=K=2*half+1
//   C/D    : VGPR r -> M = r + 8*half, N = l16
template <int NCOLS, bool RELU>
__global__ void sage_gemm_wmma(const float* __restrict__ A,
                               const float* __restrict__ B,
                               const float* __restrict__ bias,
                               float* __restrict__ C) {
    const int wave = threadIdx.x >> 5;
    const int lane = threadIdx.x & 31;
    const int half = lane >> 4;
    const int l16  = lane & 15;

    const int m0  = blockIdx.x * 16;
    const int n0  = wave * 16;
    const int row = m0 + l16;
    const int col = n0 + l16;

    const float* arow = A + (size_t)row * 128;

    v8f acc = {};
#pragma unroll
    for (int kt = 0; kt < 32; ++kt) {
        const int kk = kt * 4 + half * 2;
        const v2f a = *(const v2f*)(arow + kk);          // b64 load, 8B aligned
        v2f b;
        b[0] = B[(size_t)kk * NCOLS + col];
        b[1] = B[(size_t)(kk + 1) * NCOLS + col];
        acc = __builtin_amdgcn_wmma_f32_16x16x4_f32(
            /*neg_a=*/false, a, /*neg_b=*/false, b,
            /*c_mod=*/(short)0, acc, /*reuse_a=*/false, /*reuse_b=*/false);
    }

    const float bv = bias[col];
#pragma unroll
    for (int r = 0; r < 8; ++r) {
        const int mm = m0 + half * 8 + r;
        float v = acc[r] + bv;
        if (RELU) v = fmaxf(v, 0.0f);
        C[(size_t)mm * NCOLS + col] = v;
    }
}

// ---------------------------------------------------------------------------
extern "C" void kernel_launch(void* const* d_in, const int* in_sizes, int n_in,
                              void* d_out, int out_size, void* d_ws, size_t ws_size,
                              hipStream_t stream) {
    const float* x   = (const float*)d_in[0];
    const int*   src = (const int*)  d_in[1];
    const int*   dst = (const int*)  d_in[2];
    const float* W1  = (const float*)d_in[3];
    const float* b1  = (const float*)d_in[4];
    const float* W2  = (const float*)d_in[5];
    const float* b2  = (const float*)d_in[6];
    float* out = (float*)d_out;

    // workspace layout
    float* deg  = (float*)d_ws;
    float* bufA = deg + ((GN + 511) & ~511);        // agg / h_neigh, N x 128
    float* h1   = bufA + (size_t)GN * 128;          // layer-1 output, N x 128

    const int NF = GN * 128;                        // 12.8M
    const int T  = 256;

    // ---- degree (shared by both layers) ----
    sage_fzero<<<(GN + T - 1) / T, T, 0, stream>>>(deg, GN);
    sage_deg<<<(GE + T - 1) / T, T, 0, stream>>>(dst, deg, GE);

    // ---- layer 1 ----
    sage_fzero<<<(NF + T - 1) / T, T, 0, stream>>>(bufA, NF);
    sage_scatter<<<(GE * 32 + T - 1) / T, T, 0, stream>>>(x, src, dst, bufA, GE);
    sage_neigh<<<(NF + T - 1) / T, T, 0, stream>>>(bufA, x, deg, NF);
    // C = relu(hn @ W1 + b1) -> h1 ; 6250 row tiles x 8 col tiles (8 waves/block)
    sage_gemm_wmma<DHID, true><<<GN / 16, 32 * (DHID / 16), 0, stream>>>(bufA, W1, b1, h1);

    // ---- layer 2 ----
    sage_fzero<<<(NF + T - 1) / T, T, 0, stream>>>(bufA, NF);
    sage_scatter<<<(GE * 32 + T - 1) / T, T, 0, stream>>>(h1, src, dst, bufA, GE);
    sage_neigh<<<(NF + T - 1) / T, T, 0, stream>>>(bufA, h1, deg, NF);
    // C = hn @ W2 + b2 -> d_out ; 6250 row tiles x 4 col tiles (4 waves/block)
    sage_gemm_wmma<DOUT, false><<<GN / 16, 32 * (DOUT / 16), 0, stream>>>(bufA, W2, b2, out);
}